// SpanPairLayer_63376537420532
// MI455X (gfx1250) — compile-verified
//
#include <hip/hip_runtime.h>

#define B_    2
#define N1_   256
#define N2_   256
#define DIN_  1024
#define DRED_ 512
#define DREP_ 1024

typedef __attribute__((ext_vector_type(2))) float v2f;
typedef __attribute__((ext_vector_type(4))) float v4f;
typedef __attribute__((ext_vector_type(8))) float v8f;

// One wave computes one 16x16 fp32 tile of C = act(A @ W [+ bias]).
// A: MxK row-major, W: KxN row-major, C: MxN row-major.
// V_WMMA_F32_16X16X4_F32 fragment layout (ISA 7.12.2):
//   A 16x4: lanes 0-15 -> M=lane, VGPR{0,1} = K{0,1}; lanes 16-31 -> K{2,3}
//   B 4x16: lanes 0-15 -> N=lane, VGPR{0,1} = K{0,1}; lanes 16-31 -> K{2,3}
//   C/D:    VGPR r -> row (r + 8*(lane>=16)), col = lane&15
__global__ __launch_bounds__(128)
void wmma_gemm_f32(const float* __restrict__ A, const float* __restrict__ W,
                   const float* __restrict__ bias, float* __restrict__ C,
                   int M, int K, int N, int relu_en) {
  const int lane = threadIdx.x & 31;
  const int tn   = (blockIdx.x * 4 + threadIdx.y) * 16;  // output col base
  const int tm   = blockIdx.y * 16;                      // output row base
  const int half = lane >> 4;                            // 0: K{0,1}, 1: K{2,3}
  const int l15  = lane & 15;

  const float* Arow = A + (size_t)(tm + l15) * K + half * 2;  // contiguous K pair
  const float* Bptr = W + (size_t)(half * 2) * N + tn + l15;  // column l15, K pair rows

  v8f acc = {};
#pragma unroll 4
  for (int k = 0; k < K; k += 4) {
    v2f a = *reinterpret_cast<const v2f*>(Arow + k);   // 8B-aligned: global_load_b64
    v2f b;
    b.x = Bptr[(size_t)k * N];
    b.y = Bptr[(size_t)k * N + N];
    acc = __builtin_amdgcn_wmma_f32_16x16x4_f32(
        /*neg_a=*/false, a, /*neg_b=*/false, b,
        /*c_mod=*/(short)0, acc, /*reuse_a=*/false, /*reuse_b=*/false);
  }

  const float bcol = relu_en ? bias[tn + l15] : 0.0f;
  float* Cbase = C + (size_t)tm * N + tn + l15;
#pragma unroll
  for (int r = 0; r < 8; ++r) {
    float v = acc[r];
    if (relu_en) v = fmaxf(v + bcol, 0.0f);
    Cbase[(size_t)(r + half * 8) * N] = v;  // lanes 0-15 row r, lanes 16-31 row r+8
  }
}

// out[b,n,m,:] = relu(h1[b,n,:] + h2[b,m,:] + b_repr[:])
// 512 MiB of stores -> the dominant cost; h1/h2/bias are L2-resident.
// One block per (b,n,m); 256 threads x float4 covers DREP=1024.
__global__ __launch_bounds__(256)
void pair_add_relu(const float* __restrict__ h1, const float* __restrict__ h2,
                   const float* __restrict__ brep, float* __restrict__ out) {
  const int f = threadIdx.x;       // float4 index, 0..255
  const int m = blockIdx.x;
  const int n = blockIdx.y;
  const int b = blockIdx.z;

  const v4f* p1 = reinterpret_cast<const v4f*>(h1 + ((size_t)b * N1_ + n) * DREP_);
  const v4f* p2 = reinterpret_cast<const v4f*>(h2 + ((size_t)b * N2_ + m) * DREP_);
  const v4f* pb = reinterpret_cast<const v4f*>(brep);

  v4f x = p1[f], y = p2[f], z = pb[f];
  v4f r;
  r.x = fmaxf(x.x + y.x + z.x, 0.0f);
  r.y = fmaxf(x.y + y.y + z.y, 0.0f);
  r.z = fmaxf(x.z + y.z + z.z, 0.0f);
  r.w = fmaxf(x.w + y.w + z.w, 0.0f);

  v4f* po = reinterpret_cast<v4f*>(out + (((size_t)b * N1_ + n) * N2_ + m) * DREP_);
  __builtin_nontemporal_store(r, po + f);  // don't rinse 512MB through the caches
}

extern "C" void kernel_launch(void* const* d_in, const int* in_sizes, int n_in,
                              void* d_out, int out_size, void* d_ws, size_t ws_size,
                              hipStream_t stream) {
  const float* span1  = (const float*)d_in[0];
  const float* span2  = (const float*)d_in[1];
  const float* W_red  = (const float*)d_in[2];
  const float* b_red  = (const float*)d_in[3];
  const float* W_repr = (const float*)d_in[4];
  const float* b_rep  = (const float*)d_in[5];
  float* out = (float*)d_out;

  // Workspace layout (6 MiB total): r1 | r2 | h1 | h2
  char* ws = (char*)d_ws;
  float* r1 = (float*)(ws + (0ull << 20));
  float* r2 = (float*)(ws + (1ull << 20));
  float* h1 = (float*)(ws + (2ull << 20));
  float* h2 = (float*)(ws + (4ull << 20));

  const int M = B_ * N1_;  // 512 rows, flattened (b, n)
  dim3 blk(32, 4);         // 4 waves = 4 adjacent N-tiles per block

  // Stage 1: r = relu(span @ W_red + b_red)   (M=512, K=1024, N=512)
  wmma_gemm_f32<<<dim3(DRED_ / 64, M / 16), blk, 0, stream>>>(
      span1, W_red, b_red, r1, M, DIN_, DRED_, 1);
  wmma_gemm_f32<<<dim3(DRED_ / 64, M / 16), blk, 0, stream>>>(
      span2, W_red, b_red, r2, M, DIN_, DRED_, 1);

  // Stage 2: h1 = r1 @ Wa, h2 = r2 @ Wb       (M=512, K=512, N=1024)
  const float* Wa = W_repr;
  const float* Wb = W_repr + (size_t)DRED_ * DREP_;
  wmma_gemm_f32<<<dim3(DREP_ / 64, M / 16), blk, 0, stream>>>(
      r1, Wa, nullptr, h1, M, DRED_, DREP_, 0);
  wmma_gemm_f32<<<dim3(DREP_ / 64, M / 16), blk, 0, stream>>>(
      r2, Wb, nullptr, h2, M, DRED_, DREP_, 0);

  // Stage 3: broadcast add + bias + relu, NT-stored (bandwidth-bound).
  pair_add_relu<<<dim3(N2_, N1_, B_), 256, 0, stream>>>(h1, h2, b_rep, out);
}